// GraphSAGEEncoder_68023692034097
// MI455X (gfx1250) — compile-verified
//
#include <hip/hip_runtime.h>

typedef __attribute__((ext_vector_type(2))) float v2f;
typedef __attribute__((ext_vector_type(8))) float v8f;

constexpr int   N_NODES = 50000;
constexpr int   N_EDGES = 800000;
constexpr float BN_EPS  = 1e-5f;
constexpr float L2_EPS  = 1e-12f;

// ---------------------------------------------------------------- utilities
__global__ void zero_f32(float* __restrict__ p, long long n) {
  long long i = (long long)blockIdx.x * blockDim.x + threadIdx.x;
  long long stride = (long long)gridDim.x * blockDim.x;
  for (; i < n; i += stride) p[i] = 0.0f;
}

__global__ void degree_kernel(const int* __restrict__ dst,
                              float* __restrict__ deg, int ne) {
  int e = blockIdx.x * blockDim.x + threadIdx.x;
  if (e < ne) unsafeAtomicAdd(&deg[dst[e]], 1.0f);
}

__global__ void invert_kernel(float* __restrict__ deg, int n) {
  int i = blockIdx.x * blockDim.x + threadIdx.x;
  if (i < n) {
    float c = deg[i];
    deg[i] = (c > 0.0f) ? (1.0f / c) : 0.0f;
  }
}

// Edge-parallel scatter-add of x[src] rows into agg[dst] rows (unscaled sum).
// Channel chunk is the fastest-varying index -> coalesced loads & atomics.
template <int DI>
__global__ void scatter_sum_kernel(const float* __restrict__ x,
                                   const int* __restrict__ src,
                                   const int* __restrict__ dst,
                                   float* __restrict__ agg, int ne) {
  constexpr int C4 = DI / 4;
  long long idx = (long long)blockIdx.x * blockDim.x + threadIdx.x;
  if (idx >= (long long)ne * C4) return;
  int e = (int)(idx / C4);
  int c = (int)(idx % C4);
  int s = src[e], d = dst[e];
  const float4 v = *(const float4*)(x + (long long)s * DI + c * 4);
  float* ap = agg + (long long)d * DI + c * 4;
  unsafeAtomicAdd(ap + 0, v.x);
  unsafeAtomicAdd(ap + 1, v.y);
  unsafeAtomicAdd(ap + 2, v.z);
  unsafeAtomicAdd(ap + 3, v.w);
}

// ------------------------------------------------- fused SAGE layer (WMMA)
// h = (inv_cnt*agg) @ Wl + bl + x @ Wr ; L2-normalize rows ; BN(eval) ;
// h += PROJ ? (x @ Wp + bp) : x ; optional ReLU.
//
// Block = 64 nodes (4 M-tiles). One wave per 16-col N-tile (NT waves).
// Each wave loads its B fragments once per K-step and reuses them across
// all 4 M-tiles (register-blocked) -> 4x less L2 weight traffic.
template <int DI, int DO, bool PROJ, bool RELU>
__global__ __launch_bounds__((DO / 16) * 32) void sage_layer_kernel(
    const float* __restrict__ x,       // [N, DI]
    const float* __restrict__ agg,     // [N, DI] (unscaled neighbor sums)
    const float* __restrict__ inv_cnt, // [N]
    const float* __restrict__ Wl, const float* __restrict__ bl,
    const float* __restrict__ Wr,
    const float* __restrict__ bn_g, const float* __restrict__ bn_b,
    const float* __restrict__ bn_m, const float* __restrict__ bn_v,
    const float* __restrict__ Wp, const float* __restrict__ bp,
    float* __restrict__ out) {         // [N, DO]
  constexpr int PAD     = 4;
  constexpr int NT      = DO / 16;     // N-tiles == waves per block
  constexpr int THREADS = NT * 32;
  constexpr int MROWS   = 64;          // nodes per block (4 M-tiles)
  constexpr int TPR     = THREADS / MROWS;  // threads per row for staging

  __shared__ float aggT[MROWS][DI + PAD];
  __shared__ float xT[MROWS][DI + PAD];
  __shared__ float rowsum[MROWS];

  const int tid   = threadIdx.x;
  const int wave  = tid >> 5;
  const int lane  = tid & 31;
  const int m     = lane & 15;   // A-row-in-tile / B,C-column-in-tile
  const int hi    = lane >> 4;   // half-wave: K pair select / row+8
  const int node0 = blockIdx.x * MROWS;
  const int n0    = wave * 16;   // this wave's output-column tile

  // ---- stage 64-node tiles in LDS (TPR threads per row, float4 loads)
  {
    const int row = tid / TPR;
    const int lr  = tid % TPR;
    int gn = node0 + row;
    if (gn >= N_NODES) gn = N_NODES - 1;   // clamp tail (rows are independent)
    const float ic = inv_cnt[gn];
    const float* xrow = x   + (long long)gn * DI;
    const float* arow = agg + (long long)gn * DI;
#pragma unroll
    for (int j = 0; j < DI / (4 * TPR); ++j) {
      const int f = (lr + j * TPR) * 4;
      const float4 xv = *(const float4*)(xrow + f);
      const float4 av = *(const float4*)(arow + f);
      xT[row][f + 0] = xv.x;  xT[row][f + 1] = xv.y;
      xT[row][f + 2] = xv.z;  xT[row][f + 3] = xv.w;
      aggT[row][f + 0] = av.x * ic;  aggT[row][f + 1] = av.y * ic;
      aggT[row][f + 2] = av.z * ic;  aggT[row][f + 3] = av.w * ic;
    }
  }
  if (tid < MROWS) rowsum[tid] = 0.0f;
  __syncthreads();

  // ---- WMMA GEMMs, register-blocked over 4 M-tiles
  v8f acc[4];
  v8f accp[4];
  {
    const float blv = bl[n0 + m];
    const float bpv = PROJ ? bp[n0 + m] : 0.0f;
#pragma unroll
    for (int mt = 0; mt < 4; ++mt) {
#pragma unroll
      for (int i = 0; i < 8; ++i) acc[mt][i] = blv;
      if (PROJ) {
#pragma unroll
        for (int i = 0; i < 8; ++i) accp[mt][i] = bpv;
      }
    }
  }

#pragma unroll
  for (int k = 0; k < DI; k += 4) {
    const int kb = k + 2 * hi;
    // B fragments: loaded once, reused for all 4 M-tiles
    v2f bL; bL[0] = Wl[kb * DO + n0 + m]; bL[1] = Wl[(kb + 1) * DO + n0 + m];
    v2f bR; bR[0] = Wr[kb * DO + n0 + m]; bR[1] = Wr[(kb + 1) * DO + n0 + m];
    v2f bP;
    if (PROJ) { bP[0] = Wp[kb * DO + n0 + m]; bP[1] = Wp[(kb + 1) * DO + n0 + m]; }
#pragma unroll
    for (int mt = 0; mt < 4; ++mt) {
      const int r = mt * 16 + m;
      v2f aA; aA[0] = aggT[r][kb]; aA[1] = aggT[r][kb + 1];
      v2f aX; aX[0] = xT[r][kb];   aX[1] = xT[r][kb + 1];
      acc[mt] = __builtin_amdgcn_wmma_f32_16x16x4_f32(false, aA, false, bL,
                                                      (short)0, acc[mt],
                                                      false, false);
      acc[mt] = __builtin_amdgcn_wmma_f32_16x16x4_f32(false, aX, false, bR,
                                                      (short)0, acc[mt],
                                                      false, false);
      if (PROJ) {
        accp[mt] = __builtin_amdgcn_wmma_f32_16x16x4_f32(false, aX, false, bP,
                                                         (short)0, accp[mt],
                                                         false, false);
      }
    }
  }

  // ---- per-row sum of squares (partials over this wave's 16 columns)
  // C VGPR i = row (mt*16 + i + 8*hi), columns striped over 16 lanes.
#pragma unroll
  for (int mt = 0; mt < 4; ++mt) {
#pragma unroll
    for (int i = 0; i < 8; ++i) {
      float s = acc[mt][i] * acc[mt][i];
      s += __shfl_xor(s, 1, 16);
      s += __shfl_xor(s, 2, 16);
      s += __shfl_xor(s, 4, 16);
      s += __shfl_xor(s, 8, 16);
      if (m == 0) atomicAdd(&rowsum[mt * 16 + i + 8 * hi], s);  // ds_add_f32
    }
  }
  __syncthreads();

  // ---- epilogue: L2 norm, BN(eval), residual, ReLU, guarded store
  {
    const int col = n0 + m;
    const float g  = bn_g[col], bb = bn_b[col];
    const float mm = bn_m[col], vv = bn_v[col];
    const float bns = g * rsqrtf(vv + BN_EPS);
#pragma unroll
    for (int mt = 0; mt < 4; ++mt) {
#pragma unroll
      for (int i = 0; i < 8; ++i) {
        const int row = mt * 16 + i + 8 * hi;
        float nrm = fmaxf(sqrtf(rowsum[row]), L2_EPS);
        float h = acc[mt][i] / nrm;
        h = (h - mm) * bns + bb;
        if (PROJ) h += accp[mt][i];
        else      h += xT[row][col];  // identity residual (DI == DO)
        if (RELU) h = fmaxf(h, 0.0f);
        if (node0 + row < N_NODES)
          out[(long long)(node0 + row) * DO + col] = h;
      }
    }
  }
}

// ---------------------------------------------------------------- launcher
extern "C" void kernel_launch(void* const* d_in, const int* in_sizes, int n_in,
                              void* d_out, int out_size, void* d_ws, size_t ws_size,
                              hipStream_t stream) {
  const float* x0 = (const float*)d_in[0];
  // layer 0 (64 -> 128, proj)
  const float *L0Wl = (const float*)d_in[1],  *L0bl = (const float*)d_in[2];
  const float *L0Wr = (const float*)d_in[3];
  const float *L0g  = (const float*)d_in[4],  *L0b  = (const float*)d_in[5];
  const float *L0m  = (const float*)d_in[6],  *L0v  = (const float*)d_in[7];
  const float *L0Wp = (const float*)d_in[8],  *L0bp = (const float*)d_in[9];
  // layer 1 (128 -> 128, identity)
  const float *L1Wl = (const float*)d_in[10], *L1bl = (const float*)d_in[11];
  const float *L1Wr = (const float*)d_in[12];
  const float *L1g  = (const float*)d_in[13], *L1b  = (const float*)d_in[14];
  const float *L1m  = (const float*)d_in[15], *L1v  = (const float*)d_in[16];
  // layer 2 (128 -> 64, proj)
  const float *L2Wl = (const float*)d_in[17], *L2bl = (const float*)d_in[18];
  const float *L2Wr = (const float*)d_in[19];
  const float *L2g  = (const float*)d_in[20], *L2b  = (const float*)d_in[21];
  const float *L2m  = (const float*)d_in[22], *L2v  = (const float*)d_in[23];
  const float *L2Wp = (const float*)d_in[24], *L2bp = (const float*)d_in[25];
  const int* ei  = (const int*)d_in[26];
  const int* src = ei;
  const int* dst = ei + N_EDGES;

  float* ws  = (float*)d_ws;
  float* deg = ws;                                   // N (becomes inv_cnt)
  float* agg = deg + N_NODES;                        // N*128
  float* x1  = agg + (size_t)N_NODES * 128;          // N*128
  float* x2  = x1  + (size_t)N_NODES * 128;          // N*128
  float* y   = (float*)d_out;                        // N*64

  const int B = 256;
  const int nodeBlocks = (N_NODES + 63) / 64;        // 782 (tail guarded)

  // in-degree -> inv_cnt (in place)
  zero_f32<<<(N_NODES + B - 1) / B, B, 0, stream>>>(deg, N_NODES);
  degree_kernel<<<(N_EDGES + B - 1) / B, B, 0, stream>>>(dst, deg, N_EDGES);
  invert_kernel<<<(N_NODES + B - 1) / B, B, 0, stream>>>(deg, N_NODES);

  // ---- layer 0: 64 -> 128, proj residual, ReLU
  {
    long long n = (long long)N_NODES * 64;
    zero_f32<<<(int)((n + B - 1) / B), B, 0, stream>>>(agg, n);
    long long w = (long long)N_EDGES * (64 / 4);
    scatter_sum_kernel<64><<<(int)((w + B - 1) / B), B, 0, stream>>>(
        x0, src, dst, agg, N_EDGES);
    sage_layer_kernel<64, 128, true, true><<<nodeBlocks, 256, 0, stream>>>(
        x0, agg, deg, L0Wl, L0bl, L0Wr, L0g, L0b, L0m, L0v, L0Wp, L0bp, x1);
  }
  // ---- layer 1: 128 -> 128, identity residual, ReLU
  {
    long long n = (long long)N_NODES * 128;
    zero_f32<<<(int)((n + B - 1) / B), B, 0, stream>>>(agg, n);
    long long w = (long long)N_EDGES * (128 / 4);
    scatter_sum_kernel<128><<<(int)((w + B - 1) / B), B, 0, stream>>>(
        x1, src, dst, agg, N_EDGES);
    sage_layer_kernel<128, 128, false, true><<<nodeBlocks, 256, 0, stream>>>(
        x1, agg, deg, L1Wl, L1bl, L1Wr, L1g, L1b, L1m, L1v, nullptr, nullptr, x2);
  }
  // ---- layer 2: 128 -> 64, proj residual, no ReLU
  {
    long long n = (long long)N_NODES * 128;
    zero_f32<<<(int)((n + B - 1) / B), B, 0, stream>>>(agg, n);
    long long w = (long long)N_EDGES * (128 / 4);
    scatter_sum_kernel<128><<<(int)((w + B - 1) / B), B, 0, stream>>>(
        x2, src, dst, agg, N_EDGES);
    sage_layer_kernel<128, 64, true, false><<<nodeBlocks, 128, 0, stream>>>(
        x2, agg, deg, L2Wl, L2bl, L2Wr, L2g, L2b, L2m, L2v, L2Wp, L2bp, y);
  }
}